// ProofGNN_NextNode_15917148799635
// MI455X (gfx1250) — compile-verified
//
#include <hip/hip_runtime.h>
#include <hip/hip_bf16.h>

// ---------------------------------------------------------------------------
// ProofGNN_NextNode: 2-layer GraphSAGE(mean) + classifier, CDNA5 (gfx1250).
//
// MI455X reasoning: whole working set (~80MB) fits in 192MB L2, so the edge
// scatter (1.2GB of L2 traffic) is the bottleneck -> fp32 hardware atomics.
// GEMMs (6.8 GFLOP fp32) go through V_WMMA_F32_16X16X4_F32 (fp32 matrix path,
// wave32) with LDS-staged A tiles (stride 132 -> conflict-free ds_load_b64)
// and pre-transposed weights (single global_load_b64 per B fragment).
// ---------------------------------------------------------------------------

#define N_NODES   50000
#define N_EDGES   600000
#define EMBED     128
#define HIDDEN    128
#define NUM_TYPES 16

typedef float v2f __attribute__((ext_vector_type(2)));
typedef float v8f __attribute__((ext_vector_type(8)));

static __device__ __forceinline__ void atomic_add_f32(float* p, float v) {
  (void)__hip_atomic_fetch_add(p, v, __ATOMIC_RELAXED, __HIP_MEMORY_SCOPE_AGENT);
}

// ------------------------------- utility -----------------------------------

__global__ void fill_zero_f4(float4* __restrict__ p, int n4) {
  int i = blockIdx.x * blockDim.x + threadIdx.x;
  if (i < n4) p[i] = make_float4(0.f, 0.f, 0.f, 0.f);
}

__global__ void fill_zero_f(float* __restrict__ p, int n) {
  int i = blockIdx.x * blockDim.x + threadIdx.x;
  if (i < n) p[i] = 0.f;
}

// out[n*K + k] = in[k*N + n]   (weights are tiny; one pass, stays in L2)
__global__ void transpose_kn(const float* __restrict__ in, float* __restrict__ out,
                             int K, int N) {
  int i = blockIdx.x * blockDim.x + threadIdx.x;
  if (i < K * N) {
    int k = i / N, n = i - k * N;
    out[n * K + k] = in[i];
  }
}

// x0[i] = emb[entity[i]]  -- one wave per node, float4 per lane (128 floats)
__global__ void gather_embed(const float4* __restrict__ emb,
                             const int* __restrict__ entity,
                             float4* __restrict__ x0) {
  int wid  = (blockIdx.x * blockDim.x + threadIdx.x) >> 5;
  int lane = threadIdx.x & 31;
  if (wid >= N_NODES) return;
  int e = entity[wid];
  x0[(size_t)wid * 32 + lane] = emb[(size_t)e * 32 + lane];
}

__global__ void edge_deg(const int* __restrict__ dst, float* __restrict__ deg) {
  int e = blockIdx.x * blockDim.x + threadIdx.x;
  if (e < N_EDGES) atomic_add_f32(&deg[dst[e]], 1.0f);
}

__global__ void invert_deg(float* __restrict__ deg) {
  int i = blockIdx.x * blockDim.x + threadIdx.x;
  if (i < N_NODES) deg[i] = 1.0f / fmaxf(deg[i], 1.0f);
}

// agg[dst[e]] += x[src[e]] -- one wave per edge, 4 floats per lane.
// fp32 HW atomics: everything L2-resident on MI455X.
__global__ void scatter_accum(const int* __restrict__ src, const int* __restrict__ dst,
                              const float* __restrict__ x, float* __restrict__ agg) {
  int w    = (blockIdx.x * blockDim.x + threadIdx.x) >> 5;
  int lane = threadIdx.x & 31;
  if (w >= N_EDGES) return;
  int s = src[w], d = dst[w];
  float4 v = ((const float4*)x)[(size_t)s * 32 + lane];
  float* ap = agg + (size_t)d * 128 + lane * 4;
  atomic_add_f32(ap + 0, v.x);
  atomic_add_f32(ap + 1, v.y);
  atomic_add_f32(ap + 2, v.z);
  atomic_add_f32(ap + 3, v.w);
}

// ------------------------------ SAGE GEMM ----------------------------------
// out[m][n] = act( (agg[m]*invdeg[m]) @ WlT + b + x[m] @ WrT )
// Block = 16-node M tile; 8 waves cover N=128, each wave one 16x16 tile.
// A tiles staged in LDS (stride 132 floats: 132 mod 64 = 4 -> conflict-free
// b64 reads across the 16 rows). B fragments from transposed weights: b64.
#define ASTRIDE 132

template <int RELU>
__global__ __launch_bounds__(256)
void sage_gemm(const float* __restrict__ agg, const float* __restrict__ invdeg,
               const float* __restrict__ x,
               const float* __restrict__ WlT, const float* __restrict__ bvec,
               const float* __restrict__ WrT, float* __restrict__ out) {
  __shared__ float sA[16 * ASTRIDE];
  __shared__ float sX[16 * ASTRIDE];
  const int m0 = blockIdx.x * 16;
  const int t  = threadIdx.x;

  // stage: 2 x (16x128) tiles, 8 floats (2 float4) per thread; fold mean scale.
#pragma unroll
  for (int i = 0; i < 2; ++i) {
    int fid = t * 2 + i;          // 0..511 float4 slots
    int r   = fid >> 5;           // row 0..15
    int c4  = fid & 31;           // float4 column
    float s  = invdeg[m0 + r];
    float4 a = ((const float4*)(agg + (size_t)(m0 + r) * 128))[c4];
    a.x *= s; a.y *= s; a.z *= s; a.w *= s;
    *(float4*)&sA[r * ASTRIDE + c4 * 4] = a;
    *(float4*)&sX[r * ASTRIDE + c4 * 4] =
        ((const float4*)(x + (size_t)(m0 + r) * 128))[c4];
  }
  __syncthreads();

  const int lane = t & 31;
  const int wv   = t >> 5;        // n-tile 0..7
  const int rc   = lane & 15;     // A row (m) == B col (n) within tile
  const int kh   = lane >> 4;     // K half select (K pair = k0 + 2*kh)
  v8f acc = {};

  const float* bl = WlT + (size_t)(wv * 16 + rc) * 128;
  const float* br = WrT + (size_t)(wv * 16 + rc) * 128;

#pragma unroll 8
  for (int k0 = 0; k0 < 128; k0 += 4) {
    int kb = k0 + 2 * kh;
    v2f a = *(const v2f*)&sA[rc * ASTRIDE + kb];   // ds_load_b64
    v2f b = *(const v2f*)&bl[kb];                  // global_load_b64 (L2-hot)
    acc = __builtin_amdgcn_wmma_f32_16x16x4_f32(false, a, false, b,
                                                (short)0, acc, false, false);
  }
#pragma unroll 8
  for (int k0 = 0; k0 < 128; k0 += 4) {
    int kb = k0 + 2 * kh;
    v2f a = *(const v2f*)&sX[rc * ASTRIDE + kb];
    v2f b = *(const v2f*)&br[kb];
    acc = __builtin_amdgcn_wmma_f32_16x16x4_f32(false, a, false, b,
                                                (short)0, acc, false, false);
  }

  // C/D layout: VGPR v -> row v + 8*kh, col rc (fixed per lane)
  float bias = bvec[wv * 16 + rc];
#pragma unroll
  for (int v = 0; v < 8; ++v) {
    float val = acc[v] + bias;
    if (RELU) val = fmaxf(val, 0.f);
    out[(size_t)(m0 + v + 8 * kh) * 128 + wv * 16 + rc] = val;
  }
}

// ------------------------- classifier GEMM (N=16) --------------------------
__global__ __launch_bounds__(256)
void cls_gemm(const float* __restrict__ x, const float* __restrict__ WcT,
              const float* __restrict__ bc, float* __restrict__ out) {
  const int lane  = threadIdx.x & 31;
  const int wv    = threadIdx.x >> 5;
  const int mtile = blockIdx.x * 8 + wv;          // wave-uniform guard
  if (mtile >= (N_NODES / 16)) return;
  const int m0 = mtile * 16;
  const int rc = lane & 15;
  const int kh = lane >> 4;
  v8f acc = {};
  const float* ap = x + (size_t)(m0 + rc) * 128;
  const float* bp = WcT + (size_t)rc * 128;
#pragma unroll 8
  for (int k0 = 0; k0 < 128; k0 += 4) {
    int kb = k0 + 2 * kh;
    v2f a = *(const v2f*)&ap[kb];
    v2f b = *(const v2f*)&bp[kb];
    acc = __builtin_amdgcn_wmma_f32_16x16x4_f32(false, a, false, b,
                                                (short)0, acc, false, false);
  }
  float bias = bc[rc];
#pragma unroll
  for (int v = 0; v < 8; ++v)
    out[(size_t)(m0 + v + 8 * kh) * NUM_TYPES + rc] = acc[v] + bias;
}

// ------------------------------- launcher ----------------------------------
extern "C" void kernel_launch(void* const* d_in, const int* in_sizes, int n_in,
                              void* d_out, int out_size, void* d_ws, size_t ws_size,
                              hipStream_t stream) {
  const int*   entity = (const int*)d_in[0];
  const int*   edge   = (const int*)d_in[1];   // [2, N_EDGES]
  const float* emb    = (const float*)d_in[2];
  const float* W1l    = (const float*)d_in[3];
  const float* b1     = (const float*)d_in[4];
  const float* W1r    = (const float*)d_in[5];
  const float* W2l    = (const float*)d_in[6];
  const float* b2     = (const float*)d_in[7];
  const float* W2r    = (const float*)d_in[8];
  const float* Wc     = (const float*)d_in[9];
  const float* bc     = (const float*)d_in[10];
  float* out = (float*)d_out;
  const int* src = edge;
  const int* dst = edge + N_EDGES;

  // workspace layout (~77.3 MB total)
  float* p    = (float*)d_ws;
  float* x0   = p; p += (size_t)N_NODES * EMBED;   // features / reused as x2
  float* h1   = p; p += (size_t)N_NODES * EMBED;   // layer-1 output
  float* agg  = p; p += (size_t)N_NODES * EMBED;   // scatter accumulator
  float* invd = p; p += N_NODES;                   // deg -> 1/max(deg,1)
  float* W1lT = p; p += 128 * 128;
  float* W1rT = p; p += 128 * 128;
  float* W2lT = p; p += 128 * 128;
  float* W2rT = p; p += 128 * 128;
  float* WcT  = p; p += 128 * NUM_TYPES;
  (void)ws_size; (void)in_sizes; (void)n_in; (void)out_size;

  const int NB_FEAT4 = (N_NODES * 32 + 255) / 256;     // 6250: float4 fills
  const int NB_NODEW = N_NODES / 8;                    // 6250: wave-per-node
  const int NB_EDGEW = N_EDGES / 8;                    // 75000: wave-per-edge

  // weight transposes (tiny)
  transpose_kn<<<64, 256, 0, stream>>>(W1l, W1lT, 128, 128);
  transpose_kn<<<64, 256, 0, stream>>>(W1r, W1rT, 128, 128);
  transpose_kn<<<64, 256, 0, stream>>>(W2l, W2lT, 128, 128);
  transpose_kn<<<64, 256, 0, stream>>>(W2r, W2rT, 128, 128);
  transpose_kn<<<8, 256, 0, stream>>>(Wc, WcT, 128, NUM_TYPES);

  // degree (computed once; same graph both layers)
  fill_zero_f<<<(N_NODES + 255) / 256, 256, 0, stream>>>(invd, N_NODES);
  edge_deg<<<(N_EDGES + 255) / 256, 256, 0, stream>>>(dst, invd);
  invert_deg<<<(N_NODES + 255) / 256, 256, 0, stream>>>(invd);

  // x0 = emb[entity]
  gather_embed<<<NB_NODEW, 256, 0, stream>>>((const float4*)emb, entity, (float4*)x0);

  // layer 1: h1 = relu(mean-agg(x0) @ W1l + b1 + x0 @ W1r)
  fill_zero_f4<<<NB_FEAT4, 256, 0, stream>>>((float4*)agg, N_NODES * 32);
  scatter_accum<<<NB_EDGEW, 256, 0, stream>>>(src, dst, x0, agg);
  sage_gemm<1><<<N_NODES / 16, 256, 0, stream>>>(agg, invd, x0, W1lT, b1, W1rT, h1);

  // layer 2: x2 = mean-agg(h1) @ W2l + b2 + h1 @ W2r   (x0 reused as x2)
  fill_zero_f4<<<NB_FEAT4, 256, 0, stream>>>((float4*)agg, N_NODES * 32);
  scatter_accum<<<NB_EDGEW, 256, 0, stream>>>(src, dst, h1, agg);
  sage_gemm<0><<<N_NODES / 16, 256, 0, stream>>>(agg, invd, h1, W2lT, b2, W2rT, x0);

  // classifier: out = x2 @ Wc + bc
  cls_gemm<<<(N_NODES / 16 + 7) / 8, 256, 0, stream>>>(x0, WcT, bc, out);
}